// DenseFallbackConv3d_33818572488756
// MI455X (gfx1250) — compile-verified
//
#include <hip/hip_runtime.h>

// Problem constants (match reference)
#define BATCH 2
#define DD 96
#define HH 96
#define WW 96
#define CIN 16
#define COUT 32
#define NPTS 200000
#define NOFF 27                 // 3x3x3 taps
#define KTOT (NOFF * CIN)       // 432 reduction length
#define NCHUNK (KTOT / 4)       // 108 WMMA K-steps (K=4 per f32 WMMA)
#define APAD 436                // padded LDS row (floats), keeps 16B row alignment
#define NVOX (BATCH * DD * HH * WW)  // 1,769,472

typedef __attribute__((ext_vector_type(2))) float v2f;
typedef __attribute__((ext_vector_type(8))) float v8f;

// ---------------------------------------------------------------------------
// 1) Clear the voxel->point-index grid to -1 (7 MB of int32, int4 stores).
__global__ __launch_bounds__(256) void grid_clear_kernel(int4* __restrict__ grid4) {
    int t = blockIdx.x * blockDim.x + threadIdx.x;   // exactly NVOX/4 threads
    int4 m1; m1.x = -1; m1.y = -1; m1.z = -1; m1.w = -1;
    grid4[t] = m1;
}

// ---------------------------------------------------------------------------
// 2) Pre-swizzle weights into WMMA-B fragment order.
//    bswz[(chunk*2 + ntile)*32 + lane] = { WT[ka][n], WT[ka+1][n] }
//    ka = chunk*4 + 2*(lane>=16), n = ntile*16 + (lane&15)
//    WT[k][n] = conv_w[n][ci][o] with k = o*16 + ci  (o = tap, ci = in-chan)
__global__ __launch_bounds__(256) void wswz_kernel(const float* __restrict__ w,
                                                   v2f* __restrict__ bswz) {
    int t = blockIdx.x * blockDim.x + threadIdx.x;
    if (t >= NCHUNK * 2 * 32) return;
    int lane = t & 31;
    int tile = (t >> 5) & 1;
    int c    = t >> 6;
    int ka   = c * 4 + ((lane >> 4) << 1);
    int n    = tile * 16 + (lane & 15);
    int o0 = ka >> 4,        ci0 = ka & 15;
    int o1 = (ka + 1) >> 4,  ci1 = (ka + 1) & 15;
    v2f v;
    v.x = w[n * KTOT + ci0 * NOFF + o0];
    v.y = w[n * KTOT + ci1 * NOFF + o1];
    bswz[t] = v;
}

// ---------------------------------------------------------------------------
// 3) Scatter point ids into the dense index grid.
__global__ __launch_bounds__(256) void scatter_kernel(const int4* __restrict__ idx,
                                                      int* __restrict__ grid, int n) {
    int i = blockIdx.x * blockDim.x + threadIdx.x;
    if (i >= n) return;
    int4 v = idx[i];  // (b, z, y, x)
    int lin = ((v.x * DD + v.y) * HH + v.z) * WW + v.w;
    grid[lin] = i;
}

// ---------------------------------------------------------------------------
// 4) Main: per-wave 16-point tile -> gather 16x432 f32 into LDS -> f32 WMMA.
__global__ __launch_bounds__(64) void conv_kernel(const float* __restrict__ feats,
                                                  const int4*  __restrict__ idx,
                                                  const float* __restrict__ bias,
                                                  const int*   __restrict__ grid,
                                                  const v2f*   __restrict__ bswz,
                                                  float* __restrict__ out) {
    __shared__ __align__(16) float A_lds[2][16][APAD];

    const int lane = threadIdx.x & 31;
    const int wave = threadIdx.x >> 5;
    const int base = (blockIdx.x * 2 + wave) * 16;   // first point of this tile

    // ---- Gather: 432 (m, tap) tasks per tile; each copies 16 f32 channels.
    for (int t = lane; t < 16 * NOFF; t += 32) {
        int m = t & 15;
        int o = t >> 4;                 // tap 0..26
        int4 p = idx[base + m];         // (b, z, y, x)
        int z = p.y + (o / 9) - 1;
        int y = p.z + ((o / 3) % 3) - 1;
        int x = p.w + (o % 3) - 1;
        float4 f0 = {}, f1 = {}, f2 = {}, f3 = {};
        if ((unsigned)z < (unsigned)DD && (unsigned)y < (unsigned)HH &&
            (unsigned)x < (unsigned)WW) {
            int g = grid[((p.x * DD + z) * HH + y) * WW + x];
            if (g >= 0) {
                const float4* f = (const float4*)(feats + (size_t)g * CIN);
                f0 = f[0]; f1 = f[1]; f2 = f[2]; f3 = f[3];
            }
        }
        float4* dst = (float4*)&A_lds[wave][m][o * CIN];
        dst[0] = f0; dst[1] = f1; dst[2] = f2; dst[3] = f3;
    }
    __syncthreads();

    // ---- GEMM: out[16x32] = A[16x432] * WT[432x32], f32 WMMA, K=4 per step.
    v8f c0 = {}; v8f c1 = {};
    const int m    = lane & 15;
    const int koff = (lane >> 4) << 1;   // lanes 16..31 hold K+2, K+3
    for (int c = 0; c < NCHUNK; ++c) {
        int k0 = c * 4;
        v2f a  = *(const v2f*)&A_lds[wave][m][k0 + koff];
        v2f b0 = bswz[(c * 2 + 0) * 32 + lane];
        v2f b1 = bswz[(c * 2 + 1) * 32 + lane];
        c0 = __builtin_amdgcn_wmma_f32_16x16x4_f32(false, a, false, b0,
                                                   (short)0, c0, false, false);
        c1 = __builtin_amdgcn_wmma_f32_16x16x4_f32(false, a, false, b1,
                                                   (short)0, c1, false, false);
    }

    // ---- Epilogue: C/D layout -> out[N, 32] (+bias).
    const int n    = lane & 15;
    const int mrow = (lane >> 4) * 8;    // lanes 16..31 hold rows M+8
    float b0 = bias[n];
    float b1 = bias[16 + n];
#pragma unroll
    for (int j = 0; j < 8; ++j) {
        int prow = base + mrow + j;
        out[prow * COUT + n]       = c0[j] + b0;
        out[prow * COUT + 16 + n]  = c1[j] + b1;
    }
}

// ---------------------------------------------------------------------------
extern "C" void kernel_launch(void* const* d_in, const int* in_sizes, int n_in,
                              void* d_out, int out_size, void* d_ws, size_t ws_size,
                              hipStream_t stream) {
    const float* feats  = (const float*)d_in[0];   // [N,16] f32
    const int*   indices= (const int*)  d_in[1];   // [N,4]  i32
    const float* conv_w = (const float*)d_in[2];   // [32,16,3,3,3] f32
    const float* conv_b = (const float*)d_in[3];   // [32] f32
    float* out = (float*)d_out;                    // [N,32] f32

    int* grid = (int*)d_ws;                                        // 7,077,888 B
    v2f* bswz = (v2f*)((char*)d_ws + (size_t)NVOX * sizeof(int));  // 55,296 B

    grid_clear_kernel<<<NVOX / 4 / 256, 256, 0, stream>>>((int4*)grid);
    wswz_kernel<<<(NCHUNK * 64 + 255) / 256, 256, 0, stream>>>(conv_w, bswz);
    scatter_kernel<<<(NPTS + 255) / 256, 256, 0, stream>>>((const int4*)indices,
                                                           grid, NPTS);
    // 6250 blocks x 2 waves x 16 points = 200000 (exact -> EXEC all-1s at WMMA)
    conv_kernel<<<NPTS / 32, 64, 0, stream>>>(feats, (const int4*)indices,
                                              conv_b, grid, bswz, out);
}